// TemporalJitter_65687229825160
// MI455X (gfx1250) — compile-verified
//
#include <hip/hip_runtime.h>
#include <cstdint>
#include <cstddef>
#include <utility>

// TemporalJitter: out[b,tt,:] = x[b, src(b,tt), :] where
//   src(b,tt) = max{ t in [tt-2, tt+2] ∩ [0,T) : t + shifts[b,t] == tt }, else zeros.
// B=64, T=128, N=8192 (float32). Pure gather/copy — HBM-bandwidth bound (~512MB -> ~22us @23.3TB/s).
// gfx1250 path: async global<->LDS B128 data-mover (ASYNCcnt), one block per 32KB row,
// fold-expanded with immediate IOFFSETs (hardware adds INST_OFFSET to BOTH global and LDS addrs).

#define T_DIM 128
#define N_DIM 8192

#if defined(__HIP_DEVICE_COMPILE__) && defined(__gfx1250__) &&            \
    __has_builtin(__builtin_amdgcn_global_load_async_to_lds_b128) &&      \
    __has_builtin(__builtin_amdgcn_global_store_async_from_lds_b128)
#define USE_ASYNC_LDS 1
#else
#define USE_ASYNC_LDS 0
#endif

#if USE_ASYNC_LDS
typedef int v4i __attribute__((vector_size(16)));
typedef __attribute__((address_space(1))) char as1_char;
typedef __attribute__((address_space(3))) char as3_char;
typedef __attribute__((address_space(1))) v4i  as1_v4i;
typedef __attribute__((address_space(3))) v4i  as3_v4i;

__device__ __forceinline__ void wait_asynccnt0() {
#if __has_builtin(__builtin_amdgcn_s_wait_asynccnt)
  __builtin_amdgcn_s_wait_asynccnt(0);
#else
  asm volatile("s_wait_asynccnt 0" ::: "memory");
#endif
}

// Compile-time expansion so each IOFFSET is an integer constant expression.
template <int CHUNK, int... Ks>
__device__ __forceinline__ void async_row_load(as1_v4i* g, as3_v4i* l,
                                               std::integer_sequence<int, Ks...>) {
  (__builtin_amdgcn_global_load_async_to_lds_b128(g, l, Ks * CHUNK, 0), ...);
}
template <int CHUNK, int... Ks>
__device__ __forceinline__ void async_row_store(as1_v4i* g, as3_v4i* l,
                                                std::integer_sequence<int, Ks...>) {
  (__builtin_amdgcn_global_store_async_from_lds_b128(g, l, Ks * CHUNK, 0), ...);
}
#endif

// src(b,tt): largest t in [tt-2, tt+2] with t + shifts[b,t] == tt (shifts clipped to [-2,2])
__device__ __forceinline__ int find_src(const int* __restrict__ shifts,
                                        int b, int tt, int T) {
  int src = -1;
#pragma unroll
  for (int k = 2; k >= -2; --k) {
    const int t = tt + k;
    if (src < 0 && (unsigned)t < (unsigned)T) {
      if (t + shifts[b * T + t] == tt) src = t;
    }
  }
  return src;
}

// One block per output row (b,tt): compute src, then async-copy 32KB row via LDS.
template <int T, int N>
__global__ __launch_bounds__(256)
void TemporalJitter_fast(const float* __restrict__ x,
                         const int* __restrict__ shifts,
                         float* __restrict__ out)
{
  constexpr int NT    = 256;
  constexpr int CHUNK = NT * 16;            // bytes per step (4096)
  constexpr int STEPS = (N * 4) / CHUNK;    // 8
  static_assert(T == 128 && (N * 4) % CHUNK == 0, "shape assumptions");

  const int row = (int)blockIdx.x;          // row = b*T + tt
  const int b   = row >> 7;                 // T == 128
  const int tt  = row & (T - 1);
  const int tid = (int)threadIdx.x;

  const int src = find_src(shifts, b, tt, T);
  float* dst = out + (size_t)row * (size_t)N;

  if (src < 0) {
    float4 z; z.x = 0.f; z.y = 0.f; z.z = 0.f; z.w = 0.f;
    float4* d4 = (float4*)dst;
#pragma unroll
    for (int k = 0; k < STEPS; ++k) d4[tid + k * NT] = z;
    return;
  }

  const float* srow = x + ((size_t)b * (size_t)T + (size_t)src) * (size_t)N;

#if USE_ASYNC_LDS
  __shared__ float smem[N];                  // 32KB row staging
  as3_v4i* lb = (as3_v4i*)((as3_char*)smem + tid * 16);
  as1_v4i* gb = (as1_v4i*)((as1_char*)srow + tid * 16);
  as1_v4i* db = (as1_v4i*)((as1_char*)dst  + tid * 16);

  // 8 async B128 loads with literal offsets 0,4096,...,28672
  async_row_load<CHUNK>(gb, lb, std::make_integer_sequence<int, STEPS>{});

  wait_asynccnt0();                          // LDS written before store-from-LDS reads it

  // 8 async B128 stores, same literal offsets
  async_row_store<CHUNK>(db, lb, std::make_integer_sequence<int, STEPS>{});
  // outstanding async stores drained by implicit wait-idle at S_ENDPGM
#else
  const float4* s4 = (const float4*)srow;
  float4* d4 = (float4*)dst;
#pragma unroll
  for (int k = 0; k < STEPS; ++k) d4[tid + k * NT] = s4[tid + k * NT];
#endif
}

extern "C" void kernel_launch(void* const* d_in, const int* in_sizes, int n_in,
                              void* d_out, int out_size, void* d_ws, size_t ws_size,
                              hipStream_t stream) {
  const float* x      = (const float*)d_in[0];   // (B, T, N) float32
  const int*   shifts = (const int*)d_in[1];     // (B, T) int32
  float*       out    = (float*)d_out;           // (B, T, N) float32

  const int BT = in_sizes[1];                    // B*T = 8192 rows

  TemporalJitter_fast<T_DIM, N_DIM>
      <<<dim3((unsigned)BT), dim3(256), 0, stream>>>(x, shifts, out);
}